// SHGLNN_Task2_38165079392551
// MI455X (gfx1250) — compile-verified
//
#include <hip/hip_runtime.h>
#include <math.h>

#define NNODES  50000
#define NEDGES  800000
#define DHID    256
#define DOUT    16
#define NGRAPHS 512

typedef __attribute__((ext_vector_type(2))) float v2f;
typedef __attribute__((ext_vector_type(8))) float v8f;

// ---------------------------------------------------------------- degree / norms
__global__ void k_init_deg(float* deg) {
    int n = blockIdx.x * blockDim.x + threadIdx.x;
    if (n < NNODES) deg[n] = 1.0f;                 // +1 for self loop
}
__global__ void k_edge_deg(const int* __restrict__ dst, float* deg) {
    int e = blockIdx.x * blockDim.x + threadIdx.x;
    if (e < NEDGES) atomicAdd(&deg[dst[e]], 1.0f);
}
__global__ void k_dis(float* deg_dis) {
    int n = blockIdx.x * blockDim.x + threadIdx.x;
    if (n < NNODES) deg_dis[n] = rsqrtf(deg_dis[n]);
}
__global__ void k_norm_e(const int* __restrict__ src, const int* __restrict__ dst,
                         const float* __restrict__ dis, float* __restrict__ norme) {
    int e = blockIdx.x * blockDim.x + threadIdx.x;
    if (e < NEDGES) norme[e] = dis[src[e]] * dis[dst[e]];
}

// ---------------------------------------------------------------- WMMA f32 GEMM
// C[M x 256] = A[M x 256] * B[256 x 256], row-major.
// One wave computes a 16x64 output strip: 4 accumulators reuse each A fragment
// 4x (10 dwords/lane per 4 WMMAs) and give 4 independent WMMA dep chains.
__global__ void k_gemm_wmma(const float* __restrict__ A, const float* __restrict__ B,
                            float* __restrict__ C, int M) {
    const int wave   = blockIdx.x * (blockDim.x >> 5) + (threadIdx.x >> 5);
    const int tilesN = DHID / 64;                  // 4 strips of 64 columns
    const int tileM  = wave / tilesN;
    const int tileN  = wave % tilesN;
    if (tileM * 16 >= M) return;                   // wave-uniform: EXEC stays full

    const int lane = threadIdx.x & 31;
    const int half = lane >> 4;                    // 0: K pair {k,k+1}, 1: {k+2,k+3}
    const int l    = lane & 15;
    const int rowA = tileM * 16 + l;
    const int colB = tileN * 64 + l;               // strip base column for this lane
    const float* __restrict__ Arow = A + (size_t)rowA * DHID;

    v8f c0 = {}, c1 = {}, c2 = {}, c3 = {};
    #pragma unroll 4
    for (int k = 0; k < DHID; k += 4) {
        const int ka = k + half * 2;
        v2f a;
        a.x = Arow[ka];
        a.y = Arow[ka + 1];
        const float* __restrict__ B0 = B + (size_t)ka * DHID + colB;
        const float* __restrict__ B1 = B0 + DHID;
        v2f b0, b1, b2, b3;
        b0.x = B0[0];  b0.y = B1[0];
        b1.x = B0[16]; b1.y = B1[16];
        b2.x = B0[32]; b2.y = B1[32];
        b3.x = B0[48]; b3.y = B1[48];
        c0 = __builtin_amdgcn_wmma_f32_16x16x4_f32(false, a, false, b0, (short)0, c0, false, false);
        c1 = __builtin_amdgcn_wmma_f32_16x16x4_f32(false, a, false, b1, (short)0, c1, false, false);
        c2 = __builtin_amdgcn_wmma_f32_16x16x4_f32(false, a, false, b2, (short)0, c2, false, false);
        c3 = __builtin_amdgcn_wmma_f32_16x16x4_f32(false, a, false, b3, (short)0, c3, false, false);
    }
    // C layout per 16x16 tile: VGPR r -> row tileM*16 + half*8 + r, col = base + (lane&15)
    float* __restrict__ Crow = C + (size_t)(tileM * 16 + half * 8) * DHID + tileN * 64 + l;
    #pragma unroll
    for (int r = 0; r < 8; ++r) {
        Crow[(size_t)r * DHID + 0]  = c0[r];
        Crow[(size_t)r * DHID + 16] = c1[r];
        Crow[(size_t)r * DHID + 32] = c2[r];
        Crow[(size_t)r * DHID + 48] = c3[r];
    }
}

// ---------------------------------------------------------------- GCN aggregation
// agg[n,j] = ht[n,j] * dis[n]^2 + bias[j]   (self-loop term + bias)
__global__ void k_init_agg(const float* __restrict__ ht, const float* __restrict__ dis,
                           const float* __restrict__ bias, float* __restrict__ agg) {
    int n = blockIdx.x, j = threadIdx.x;
    float d = dis[n];
    agg[(size_t)n * DHID + j] = ht[(size_t)n * DHID + j] * (d * d) + bias[j];
}
// agg[dst,:] += ht[src,:] * norm_e   (64 threads/edge, float4 per thread)
__global__ void k_scatter(const float* __restrict__ ht, const int* __restrict__ src,
                          const int* __restrict__ dst, const float* __restrict__ norme,
                          float* __restrict__ agg) {
    int gid = blockIdx.x * blockDim.x + threadIdx.x;   // NEDGES*64 < 2^31
    int e = gid >> 6;
    if (e >= NEDGES) return;
    int c = (gid & 63) << 2;
    float w = norme[e];
    const float4 v = *(const float4*)(ht + (size_t)src[e] * DHID + c);
    float* a = agg + (size_t)dst[e] * DHID + c;
    atomicAdd(a + 0, v.x * w);
    atomicAdd(a + 1, v.y * w);
    atomicAdd(a + 2, v.z * w);
    atomicAdd(a + 3, v.w * w);
}

// ---------------------------------------------------------------- attn prep
// In-place ReLU; logit[n] = h.attn_W + attn_b ; rowsum[n] = h.hw
__global__ void k_relu_dot(float* __restrict__ h, const float* __restrict__ aW,
                           const float* __restrict__ ab, const float* __restrict__ hw,
                           float* __restrict__ logit, float* __restrict__ rowsum) {
    int n = blockIdx.x * (blockDim.x >> 5) + (threadIdx.x >> 5);
    int lane = threadIdx.x & 31;
    if (n >= NNODES) return;
    float a1 = 0.f, a2 = 0.f;
    #pragma unroll
    for (int j = lane; j < DHID; j += 32) {
        float v = h[(size_t)n * DHID + j];
        v = fmaxf(v, 0.f);
        h[(size_t)n * DHID + j] = v;
        a1 += v * aW[j];
        a2 += v * hw[j];
    }
    #pragma unroll
    for (int o = 16; o > 0; o >>= 1) {
        a1 += __shfl_down(a1, o, 32);
        a2 += __shfl_down(a2, o, 32);
    }
    if (lane == 0) { logit[n] = a1 + ab[0]; rowsum[n] = a2; }
}

__global__ void k_max(const float* __restrict__ logit, float* red) {
    __shared__ float sm[1024];
    float m = -INFINITY;
    for (int i = threadIdx.x; i < NNODES; i += blockDim.x) m = fmaxf(m, logit[i]);
    sm[threadIdx.x] = m; __syncthreads();
    for (int s = blockDim.x >> 1; s > 0; s >>= 1) {
        if (threadIdx.x < s) sm[threadIdx.x] = fmaxf(sm[threadIdx.x], sm[threadIdx.x + s]);
        __syncthreads();
    }
    if (threadIdx.x == 0) red[0] = sm[0];
}
__global__ void k_sumexp(const float* __restrict__ logit, float* red) {
    __shared__ float sm[1024];
    float m = red[0], s = 0.f;
    for (int i = threadIdx.x; i < NNODES; i += blockDim.x) s += expf(logit[i] - m);
    sm[threadIdx.x] = s; __syncthreads();
    for (int k = blockDim.x >> 1; k > 0; k >>= 1) {
        if (threadIdx.x < k) sm[threadIdx.x] += sm[threadIdx.x + k];
        __syncthreads();
    }
    if (threadIdx.x == 0) red[1] = sm[0];
}

// s1[n] = softmax(logit)[n] * rowsum[n]; sagg[n] = s1[n]*dis[n]^2 (self term)
__global__ void k_attn_scalar(const float* __restrict__ logit, const float* __restrict__ rowsum,
                              const float* __restrict__ red, const float* __restrict__ dis,
                              float* __restrict__ s1, float* __restrict__ sagg) {
    int n = blockIdx.x * blockDim.x + threadIdx.x;
    if (n >= NNODES) return;
    float a = expf(logit[n] - red[0]) / red[1];
    float v = a * rowsum[n];
    s1[n] = v;
    float d = dis[n];
    sagg[n] = v * d * d;
}
__global__ void k_scatter_scalar(const float* __restrict__ s1, const int* __restrict__ src,
                                 const int* __restrict__ dst, const float* __restrict__ norme,
                                 float* __restrict__ sagg) {
    int e = blockIdx.x * blockDim.x + threadIdx.x;
    if (e < NEDGES) atomicAdd(&sagg[dst[e]], s1[src[e]] * norme[e]);
}
// h2[n,j] = relu(sagg[n]*W1[j] + b1[j])  (layer-1 GCN collapsed to outer product)
__global__ void k_outer_relu(const float* __restrict__ sagg, const float* __restrict__ W1,
                             const float* __restrict__ b1, float* __restrict__ h2) {
    int n = blockIdx.x, j = threadIdx.x;
    h2[(size_t)n * DHID + j] = fmaxf(sagg[n] * W1[j] + b1[j], 0.f);
}

// ---------------------------------------------------------------- pooling / head
__global__ void k_zero_pool(float* pool, float* cnt) {
    int i = blockIdx.x * blockDim.x + threadIdx.x;
    if (i < NGRAPHS * DHID) pool[i] = 0.f;
    if (i < NGRAPHS) cnt[i] = 0.f;
}
__global__ void k_relu_pool(const float* __restrict__ agg, const int* __restrict__ batch,
                            float* __restrict__ pool, float* __restrict__ cnt) {
    int n = blockIdx.x, j = threadIdx.x;
    float v = fmaxf(agg[(size_t)n * DHID + j], 0.f);
    int g = batch[n];
    atomicAdd(&pool[(size_t)g * DHID + j], v);
    if (j == 0) atomicAdd(&cnt[g], 1.0f);
}
// One wave per graph: logits = pooled @ W_out + b_out, then log_softmax over 16.
__global__ void k_final(const float* __restrict__ pool, const float* __restrict__ cnt,
                        const float* __restrict__ Wout, const float* __restrict__ bout,
                        float* __restrict__ out) {
    int g = blockIdx.x * (blockDim.x >> 5) + (threadIdx.x >> 5);
    int lane = threadIdx.x & 31;
    if (g >= NGRAPHS) return;
    int k = lane & 15;                               // duplicated across lane halves
    float inv = 1.0f / fmaxf(cnt[g], 1.0f);
    float acc = bout[k];
    for (int j = 0; j < DHID; ++j)
        acc += pool[(size_t)g * DHID + j] * inv * Wout[j * DOUT + k];
    float m = acc;
    #pragma unroll
    for (int o = 1; o < 16; o <<= 1) m = fmaxf(m, __shfl_xor(m, o, 32));
    float s = expf(acc - m);
    #pragma unroll
    for (int o = 1; o < 16; o <<= 1) s += __shfl_xor(s, o, 32);
    float r = acc - m - logf(s);
    if (lane < 16) out[(size_t)g * DOUT + lane] = r;
}

// ---------------------------------------------------------------- launch
extern "C" void kernel_launch(void* const* d_in, const int* in_sizes, int n_in,
                              void* d_out, int out_size, void* d_ws, size_t ws_size,
                              hipStream_t stream) {
    const float* x      = (const float*)d_in[0];
    const int*   ei     = (const int*)d_in[1];
    const int*   src    = ei;
    const int*   dst    = ei + NEDGES;
    const int*   batch  = (const int*)d_in[2];
    const float* W0     = (const float*)d_in[3];
    const float* b0     = (const float*)d_in[4];
    const float* attn_W = (const float*)d_in[5];
    const float* attn_b = (const float*)d_in[6];
    const float* hw     = (const float*)d_in[7];
    const float* W1     = (const float*)d_in[8];
    const float* b1     = (const float*)d_in[9];
    const float* W2     = (const float*)d_in[10];
    const float* b2     = (const float*)d_in[11];
    const float* Wout   = (const float*)d_in[12];
    const float* bout   = (const float*)d_in[13];
    float* out = (float*)d_out;

    char* ws = (char*)d_ws;
    size_t off = 0;
    auto alloc = [&](size_t bytes) -> float* {
        float* p = (float*)(ws + off);
        off = (off + bytes + 255) & ~(size_t)255;
        return p;
    };
    float* bufA  = alloc((size_t)NNODES * DHID * 4);   // GEMM out / agg ping-pong
    float* bufB  = alloc((size_t)NNODES * DHID * 4);
    float* dis   = alloc((size_t)NNODES * 4);          // deg -> deg^-1/2 in place
    float* norme = alloc((size_t)NEDGES * 4);
    float* logit = alloc((size_t)NNODES * 4);
    float* rowsum= alloc((size_t)NNODES * 4);
    float* s1    = alloc((size_t)NNODES * 4);
    float* sagg  = alloc((size_t)NNODES * 4);
    float* pool  = alloc((size_t)NGRAPHS * DHID * 4);
    float* cnt   = alloc((size_t)NGRAPHS * 4);
    float* red   = alloc(256);

    const int TB = 256;
    const int nodeB   = (NNODES + TB - 1) / TB;                    // 196
    const int edgeB   = (NEDGES + TB - 1) / TB;                    // 3125
    const int gemmWaves = (NNODES / 16) * (DHID / 64);             // 12500
    const int gemmB   = (gemmWaves + (TB / 32) - 1) / (TB / 32);   // 1563
    const int waveB   = (NNODES + (TB / 32) - 1) / (TB / 32);      // 6250
    const int scatB   = (NEDGES * 64 + TB - 1) / TB;               // 200000

    // degrees & normalization
    k_init_deg<<<nodeB, TB, 0, stream>>>(dis);
    k_edge_deg<<<edgeB, TB, 0, stream>>>(dst, dis);
    k_dis<<<nodeB, TB, 0, stream>>>(dis);
    k_norm_e<<<edgeB, TB, 0, stream>>>(src, dst, dis, norme);

    // ---- layer 0: h = relu(GCN(x, W0, b0))
    k_gemm_wmma<<<gemmB, TB, 0, stream>>>(x, W0, bufA, NNODES);
    k_init_agg<<<NNODES, DHID, 0, stream>>>(bufA, dis, b0, bufB);
    k_scatter<<<scatB, TB, 0, stream>>>(bufA, src, dst, norme, bufB);
    k_relu_dot<<<waveB, TB, 0, stream>>>(bufB, attn_W, attn_b, hw, logit, rowsum);

    // ---- attention softmax over nodes -> scalar per node
    k_max<<<1, 1024, 0, stream>>>(logit, red);
    k_sumexp<<<1, 1024, 0, stream>>>(logit, red);
    k_attn_scalar<<<nodeB, TB, 0, stream>>>(logit, rowsum, red, dis, s1, sagg);

    // ---- layer 1 (rank-1): scalar aggregation then outer product
    k_scatter_scalar<<<edgeB, TB, 0, stream>>>(s1, src, dst, norme, sagg);
    k_outer_relu<<<NNODES, DHID, 0, stream>>>(sagg, W1, b1, bufA);

    // ---- layer 2: h = relu(GCN(h2, W2, b2))
    k_gemm_wmma<<<gemmB, TB, 0, stream>>>(bufA, W2, bufB, NNODES);
    k_init_agg<<<NNODES, DHID, 0, stream>>>(bufB, dis, b2, bufA);
    k_scatter<<<scatB, TB, 0, stream>>>(bufB, src, dst, norme, bufA);

    // ---- mean pool + classifier head
    k_zero_pool<<<(NGRAPHS * DHID + TB - 1) / TB, TB, 0, stream>>>(pool, cnt);
    k_relu_pool<<<NNODES, DHID, 0, stream>>>(bufA, batch, pool, cnt);
    k_final<<<(NGRAPHS + (TB / 32) - 1) / (TB / 32), TB, 0, stream>>>(pool, cnt, Wout, bout, out);
}